// Graph_13726715478523
// MI455X (gfx1250) — compile-verified
//
#include <hip/hip_runtime.h>
#include <math.h>

#define NNODES 10000
#define NEDGES 160000
#define BDIM   8
#define DDIM   64
#define BD     (BDIM * DDIM)            // 512 elements per (node|edge)
#define NBD    (NNODES * BD)            // 5,120,000
#define ETOT   (NEDGES * BD)            // 81,920,000

typedef float v2f __attribute__((ext_vector_type(2)));
typedef float v8f __attribute__((ext_vector_type(8)));

// ---------------------------------------------------------------------------
// Kernel 1: P = state @ W[0:64,:],  Q = state @ W[64:128,:]
// state viewed as [80000 x 64] row-major; W is [128 x 64] row-major.
// One wave32 computes one 16x16 tile of BOTH P and Q (A reuse), K in steps of 4
// via V_WMMA_F32_16X16X4_F32.
// ---------------------------------------------------------------------------
__global__ __launch_bounds__(256) void gemm_pq(const float* __restrict__ S,
                                               const float* __restrict__ W,
                                               float* __restrict__ P,
                                               float* __restrict__ Q) {
    const int wave = threadIdx.x >> 5;
    const int lane = threadIdx.x & 31;
    const int tile = blockIdx.x * 8 + wave;   // 20000 tiles total
    const int mt = tile >> 2;                 // 5000 tiles along M (80000/16)
    const int nt = tile & 3;                  // 4 tiles along N (64/16)
    const int m0 = mt * 16;
    const int n0 = nt * 16;

    const int lo = lane & 15;                 // A: row index, B/D: col index
    const int hi = lane >> 4;                 // half-wave select
    const int kb = hi * 2;                    // A/B k sub-offset per ISA layout

    v8f accP = {};
    v8f accQ = {};

    const float* __restrict__ Srow = S + (size_t)(m0 + lo) * DDIM;

    for (int k = 0; k < DDIM; k += 4) {
        // A 16x4 f32: v0 = K = kb, v1 = K = kb+1 (per half-wave)
        v2f a;
        a.x = Srow[k + kb];
        a.y = Srow[k + kb + 1];

        const int col = n0 + lo;
        // B 4x16 f32 tiles from W (W0 = rows [0,64), W1 = rows [64,128))
        v2f b0, b1;
        b0.x = W[(k + kb) * DDIM + col];
        b0.y = W[(k + kb + 1) * DDIM + col];
        b1.x = W[(64 + k + kb) * DDIM + col];
        b1.y = W[(64 + k + kb + 1) * DDIM + col];

        accP = __builtin_amdgcn_wmma_f32_16x16x4_f32(
            false, a, false, b0, (short)0, accP, false, false);
        accQ = __builtin_amdgcn_wmma_f32_16x16x4_f32(
            false, a, false, b1, (short)0, accQ, false, false);
    }

    // D layout: lane holds col = lane&15; VGPR v holds row v + 8*(lane>>4)
#pragma unroll
    for (int v = 0; v < 8; ++v) {
        const int m = m0 + v + 8 * hi;
        const int n = n0 + lo;
        P[(size_t)m * DDIM + n] = accP[v];
        Q[(size_t)m * DDIM + n] = accQ[v];
    }
}

// ---------------------------------------------------------------------------
// Init: amax = -inf, denom = 0, out = 0 (must run every launch; atomics follow)
// ---------------------------------------------------------------------------
__global__ void init_k(float* __restrict__ amax, float* __restrict__ denom,
                       float* __restrict__ out) {
    const int i = blockIdx.x * blockDim.x + threadIdx.x;
    if (i < NBD) {
        amax[i]  = -INFINITY;
        denom[i] = 0.0f;
        out[i]   = 0.0f;
    }
}

__device__ __forceinline__ float leaky02(float x) {
    return x > 0.0f ? x : 0.2f * x;
}

// Float atomic max without CAS: int-max for non-negative, uint-min for negative.
// Valid with -inf initialization (monotone IEEE bit ordering trick).
__device__ __forceinline__ void atomic_max_f32(float* addr, float v) {
    if (v >= 0.0f)
        atomicMax((int*)addr, __float_as_int(v));
    else
        atomicMin((unsigned int*)addr, __float_as_uint(v));
}

__device__ __forceinline__ float edge_alpha(const float* __restrict__ P,
                                            const float* __restrict__ Q,
                                            int s, int d, int j, float w) {
    return leaky02(P[(size_t)s * BD + j] + Q[(size_t)d * BD + j]) * w;
}

// ---------------------------------------------------------------------------
// Pass 1: segment max of alpha per dst element
// ---------------------------------------------------------------------------
__global__ void edge_max_k(const float* __restrict__ P, const float* __restrict__ Q,
                           const int* __restrict__ src, const int* __restrict__ dst,
                           const float* __restrict__ dist, float* __restrict__ amax) {
    const int idx = blockIdx.x * blockDim.x + threadIdx.x;
    if (idx >= ETOT) return;
    const int e = idx >> 9;      // edge (block-uniform: 512 % 256 == 0)
    const int j = idx & 511;     // element within [B,D]
    const int s = src[e], d = dst[e];
    const float a = edge_alpha(P, Q, s, d, j, dist[e]);
    atomic_max_f32(&amax[(size_t)d * BD + j], a);
}

// ---------------------------------------------------------------------------
// Pass 2: denom = segment sum of exp(alpha - amax[dst])
// ---------------------------------------------------------------------------
__global__ void edge_sum_k(const float* __restrict__ P, const float* __restrict__ Q,
                           const int* __restrict__ src, const int* __restrict__ dst,
                           const float* __restrict__ dist,
                           const float* __restrict__ amax, float* __restrict__ denom) {
    const int idx = blockIdx.x * blockDim.x + threadIdx.x;
    if (idx >= ETOT) return;
    const int e = idx >> 9;
    const int j = idx & 511;
    const int s = src[e], d = dst[e];
    const size_t dj = (size_t)d * BD + j;
    const float a  = edge_alpha(P, Q, s, d, j, dist[e]);
    const float ex = expf(a - amax[dj]);
    atomicAdd(&denom[dj], ex);
}

// ---------------------------------------------------------------------------
// Pass 3: out[dst] += (exp(alpha - amax)/denom) * state[src]
// ---------------------------------------------------------------------------
__global__ void edge_acc_k(const float* __restrict__ P, const float* __restrict__ Q,
                           const float* __restrict__ state,
                           const int* __restrict__ src, const int* __restrict__ dst,
                           const float* __restrict__ dist,
                           const float* __restrict__ amax,
                           const float* __restrict__ denom,
                           float* __restrict__ out) {
    const int idx = blockIdx.x * blockDim.x + threadIdx.x;
    if (idx >= ETOT) return;
    const int e = idx >> 9;
    const int j = idx & 511;
    const int s = src[e], d = dst[e];
    const size_t dj = (size_t)d * BD + j;
    const float a    = edge_alpha(P, Q, s, d, j, dist[e]);
    const float ex   = expf(a - amax[dj]);
    const float attn = ex / denom[dj];
    atomicAdd(&out[dj], attn * state[(size_t)s * BD + j]);
}

// ---------------------------------------------------------------------------
// Final: relu in place on the accumulated output
// ---------------------------------------------------------------------------
__global__ void relu_k(float* __restrict__ out) {
    const int i = blockIdx.x * blockDim.x + threadIdx.x;
    if (i < NBD) out[i] = fmaxf(out[i], 0.0f);
}

extern "C" void kernel_launch(void* const* d_in, const int* in_sizes, int n_in,
                              void* d_out, int out_size, void* d_ws, size_t ws_size,
                              hipStream_t stream) {
    const float* state  = (const float*)d_in[0];
    // d_in[1] = feature: unused by the GAT math
    const float* weight = (const float*)d_in[2];
    const int*   src    = (const int*)d_in[3];
    const int*   dst    = (const int*)d_in[4];
    const float* dist   = (const float*)d_in[5];
    float*       out    = (float*)d_out;

    // Workspace layout: P | Q | amax | denom  (4 * 20.48 MB = 81.9 MB)
    float* P     = (float*)d_ws;
    float* Q     = P + NBD;
    float* amax  = Q + NBD;
    float* denom = amax + NBD;

    // 20000 16x16 tiles (both P and Q per tile), 8 waves per block
    gemm_pq<<<2500, 256, 0, stream>>>(state, weight, P, Q);

    init_k<<<(NBD + 255) / 256, 256, 0, stream>>>(amax, denom, out);

    const int eblocks = (ETOT + 255) / 256;
    edge_max_k<<<eblocks, 256, 0, stream>>>(P, Q, src, dst, dist, amax);
    edge_sum_k<<<eblocks, 256, 0, stream>>>(P, Q, src, dst, dist, amax, denom);
    edge_acc_k<<<eblocks, 256, 0, stream>>>(P, Q, state, src, dst, dist, amax, denom, out);

    relu_k<<<(NBD + 255) / 256, 256, 0, stream>>>(out);
}